// CausalSelfAttentionKV_28028956574320
// MI455X (gfx1250) — compile-verified
//
#include <hip/hip_runtime.h>
#include <hip/hip_bf16.h>

// ---------------------------------------------------------------------------
// CausalSelfAttention with KV cache, fp32, gfx1250 (MI455X).
//
// Roofline: past KV = 537MB must be read once and the concatenated KV output
// (537MB) written once -> ~1.07GB -> ~46us floor at 23.3 TB/s. To hit that
// floor the past->output KV copy is FUSED into the attention kernel: each
// K tile / V chunk is copied (coalesced b128) immediately before being
// consumed as WMMA fragments, so the second read hits WGP$/L2, not HBM.
// All matmuls use V_WMMA_F32_16X16X4_F32 (full fp32 on the matrix pipe).
// Per-head scores (8 x 4104 = 128KB) live in the 320KB WGP LDS.
// ---------------------------------------------------------------------------

typedef __attribute__((ext_vector_type(2))) float v2f;
typedef __attribute__((ext_vector_type(8))) float v8f;

__device__ __forceinline__ v8f wmma_f32_16x16x4(v2f a, v2f b, v8f c) {
  // D(16x16) = A(16x4) * B(4x16) + C   (fp32, wave32)
  return __builtin_amdgcn_wmma_f32_16x16x4_f32(
      /*neg_a=*/false, a, /*neg_b=*/false, b,
      /*c_mod=*/(short)0, c, /*reuse_a=*/false, /*reuse_b=*/false);
}

// Problem constants (fixed by the reference)
constexpr int B_  = 16;
constexpr int Tn  = 8;       // T_new
constexpr int C_  = 1024;
constexpr int H_  = 16;
constexpr int HD  = 64;      // head dim
constexpr int Tp  = 4096;    // T_past
constexpr int Tf  = Tp + Tn; // 4104
constexpr int QKV_COLS = 3 * C_;

// ---------------------------------------------------------------------------
// Kernel 1: qkv = x @ W_attn^T + b_attn ; scatter q (scaled by 1/8) to ws,
// k_new/v_new into the appended rows of the output KV cache.
// One wave per 16x16 output tile.
// ---------------------------------------------------------------------------
__global__ __launch_bounds__(256) void qkv_gemm(
    const float* __restrict__ x, const float* __restrict__ W,
    const float* __restrict__ bias, float* __restrict__ q_ws,
    float* __restrict__ out_k, float* __restrict__ out_v) {
  const int lane = threadIdx.x & 31;
  const int wave = threadIdx.x >> 5;
  const int tile = blockIdx.x * 8 + wave;          // 8 row-tiles * 192 col-tiles
  const int nColTiles = QKV_COLS / 16;             // 192
  const int rowBase = (tile / nColTiles) * 16;
  const int colBase = (tile % nColTiles) * 16;

  const float* arow = x + (size_t)(rowBase + (lane & 15)) * C_ + (lane >> 4) * 2;
  const float* brow = W + (size_t)(colBase + (lane & 15)) * C_ + (lane >> 4) * 2;

  v8f acc = {};
#pragma unroll 8
  for (int kk = 0; kk < C_; kk += 4) {
    v2f a = *(const v2f*)(arow + kk);
    v2f b = *(const v2f*)(brow + kk);
    acc = wmma_f32_16x16x4(a, b, acc);
  }

  const int n = lane & 15;
  const int mAdd = (lane >> 4) * 8;
#pragma unroll
  for (int i = 0; i < 8; ++i) {
    const int r = rowBase + i + mAdd;              // global row = b*8 + t
    const int c = colBase + n;                     // 0..3071
    const int bb = r >> 3, tt = r & 7;
    float val = acc[i] + bias[c];
    if (c < C_) {                                  // q -> workspace, fold 1/sqrt(hd)
      const int h = c >> 6, d = c & 63;
      q_ws[((size_t)(bb * H_ + h) * Tn + tt) * HD + d] = val * 0.125f;
    } else if (c < 2 * C_) {                       // k_new -> appended cache rows
      const int cc = c - C_, h = cc >> 6, d = cc & 63;
      out_k[((size_t)(bb * H_ + h) * Tf + (Tp + tt)) * HD + d] = val;
    } else {                                       // v_new
      const int cc = c - 2 * C_, h = cc >> 6, d = cc & 63;
      out_v[((size_t)(bb * H_ + h) * Tf + (Tp + tt)) * HD + d] = val;
    }
  }
}

// ---------------------------------------------------------------------------
// Kernel 2: fused KV-copy + attention. One workgroup (512 thr = 16 waves)
// per (b,h).
//  Phase 1: per 16-key tile: copy past_k tile -> out_k (b128), then
//           S = Qs @ K^T via WMMA (fragments read past_k -> cache hits);
//           scores to LDS with causal mask. Last tile reads out_k (new rows).
//  Phase 2: per-row softmax in LDS (wave per row, wave32 shuffles).
//  Phase 3: Y = P @ V. 16 waves = 4 hd col-tiles x 4 strided K-groups.
//           colTile-0 waves copy each 4-row V chunk past_v -> out_v exactly
//           once; fragments read past_v/out_v. Partials merged in LDS.
// ---------------------------------------------------------------------------
__global__ __launch_bounds__(512) void attn_fused(
    const float* __restrict__ q_ws,
    const float* __restrict__ past_k, const float* __restrict__ past_v,
    float* __restrict__ out_k, float* __restrict__ out_v,
    float* __restrict__ y_ws) {
  extern __shared__ float smem[];
  float* s_scores = smem;                 // 8 * 4104
  float* s_q      = s_scores + 8 * Tf;    // 16 * 64 (rows 8..15 zero)
  float* s_part   = s_q + 16 * HD;        // 16 waves * 256
  float* s_recip  = s_part + 16 * 256;    // 8

  const int lane = threadIdx.x & 31;
  const int wave = threadIdx.x >> 5;
  const int bh   = blockIdx.x;            // b*H + h
  const float* qh  = q_ws   + (size_t)bh * Tn * HD;
  const float* pkh = past_k + (size_t)bh * Tp * HD;
  const float* pvh = past_v + (size_t)bh * Tp * HD;
  float* okh = out_k + (size_t)bh * Tf * HD;
  float* ovh = out_v + (size_t)bh * Tf * HD;

  // Q tile into LDS, zero-padded to 16 rows (q already carries 1/sqrt(hd)).
  for (int i = threadIdx.x; i < 16 * HD; i += blockDim.x) {
    const int m = i >> 6, d = i & 63;
    s_q[i] = (m < Tn) ? qh[m * HD + d] : 0.0f;
  }
  __syncthreads();

  // ---- Phase 1: copy K tile, then scores ----
  const int nTiles = (Tf + 15) / 16;      // 257 (last tile: 8 new keys)
  for (int t = wave; t < nTiles; t += 16) {
    const int kb = t * 16;
    const bool isPast = (kb < Tp);        // wave-uniform
    if (isPast) {
      // coalesced 4KB tile copy past_k -> out_k (also warms WGP$/L2)
      const float4* s4 = (const float4*)(pkh + (size_t)kb * HD);
      float4* d4 = (float4*)(okh + (size_t)kb * HD);
#pragma unroll
      for (int j = 0; j < 8; ++j) d4[lane + 32 * j] = s4[lane + 32 * j];
      // prefetch this wave's next tile (global_prefetch_b8)
      const int kbn = kb + 16 * 16;
      if (kbn < Tp)
        __builtin_prefetch(pkh + (size_t)kbn * HD + lane * 32, 0, 0);
    }
    const float* ksrc = isPast ? pkh : okh;   // new rows come from out_k
    const float* kbase = ksrc + (size_t)(kb + (lane & 15)) * HD + (lane >> 4) * 2;
    const float* qbase = s_q + (lane & 15) * HD + (lane >> 4) * 2;
    v8f acc = {};
#pragma unroll
    for (int d = 0; d < HD; d += 4) {     // 16 WMMA per key tile
      v2f a = *(const v2f*)(qbase + d);   // A[m][d..]  m = lane&15
      v2f b = *(const v2f*)(kbase + d);   // B[d][n] = K[kb+n][d]
      acc = wmma_f32_16x16x4(a, b, acc);
    }
    // rows 0..7 live in lanes 0..15 (slot i = row i); lanes 16..31 are padding
    if (lane < 16) {
      const int kidx = kb + lane;         // n == lane for lanes 0..15
      if (kidx < Tf) {
#pragma unroll
        for (int i = 0; i < 8; ++i) {
          float s = acc[i];
          if (kidx - Tp > i) s = -1e30f;  // causal mask (query pos Tp + i)
          s_scores[i * Tf + kidx] = s;
        }
      }
    }
  }
  __syncthreads();

  // ---- Phase 2: softmax, one wave per query row (waves 0..7) ----
  if (wave < 8) {
    const int row = wave;
    float* srow = s_scores + row * Tf;
    float mx = -1e30f;
    for (int j = lane; j < Tf; j += 32) mx = fmaxf(mx, srow[j]);
#pragma unroll
    for (int o = 16; o >= 1; o >>= 1) mx = fmaxf(mx, __shfl_xor(mx, o, 32));
    float sum = 0.0f;
    for (int j = lane; j < Tf; j += 32) {
      const float e = __expf(srow[j] - mx);
      srow[j] = e;
      sum += e;
    }
#pragma unroll
    for (int o = 16; o >= 1; o >>= 1) sum += __shfl_xor(sum, o, 32);
    if (lane == 0) s_recip[row] = 1.0f / sum;
  }
  __syncthreads();

  // ---- Phase 3: Y = P @ V, fused V copy ----
  const int colTile = wave & 3;           // hd tile: n in [colTile*16, +16)
  const int kq      = wave >> 2;          // strided K-chunk group (mod 4)
  const int m       = lane & 15;
  v8f acc = {};
  const float* arow = s_scores + (size_t)m * Tf + (lane >> 4) * 2;
  const int nOff = colTile * 16 + m;
  const int nChunks = Tf / 4;             // 1026 chunks of 4 key rows
  for (int c = kq; c < nChunks; c += 4) {
    const int k = 4 * c;
    const bool isPast = (k < Tp);         // chunk-aligned: uniform per chunk
    if (colTile == 0 && isPast) {
      // copy this 1KB V chunk past_v -> out_v exactly once (warms caches)
      const float4* s4 = (const float4*)(pvh + (size_t)k * HD);
      float4* d4 = (float4*)(ovh + (size_t)k * HD);
      d4[lane] = s4[lane];
      d4[lane + 32] = s4[lane + 32];
    }
    const float* vsrc = isPast ? pvh : ovh;
    v2f a = {};
    if (m < 8) a = *(const v2f*)(arow + k);     // padding rows -> 0
    const int kr = k + (lane >> 4) * 2;
    v2f b;
    b.x = vsrc[(size_t)kr * HD + nOff];         // B[k][n] = V[k][n]
    b.y = vsrc[(size_t)(kr + 1) * HD + nOff];
    acc = wmma_f32_16x16x4(a, b, acc);
  }
  // merge the 4 K-group partials per column tile in LDS
  {
    float* pp = s_part + wave * 256;
#pragma unroll
    for (int i = 0; i < 8; ++i) pp[i * 32 + lane] = acc[i];
  }
  __syncthreads();
  if (wave < 4 && lane < 16) {
    const int ct = wave;
    const int bb = bh >> 4, hh = bh & 15;
#pragma unroll
    for (int i = 0; i < 8; ++i) {               // row i, col lane
      float val = 0.0f;
#pragma unroll
      for (int p = 0; p < 4; ++p)
        val += s_part[(size_t)(ct + 4 * p) * 256 + i * 32 + lane];
      val *= s_recip[i];
      y_ws[(size_t)(bb * Tn + i) * C_ + hh * HD + ct * 16 + lane] = val;
    }
  }
}

// ---------------------------------------------------------------------------
// Kernel 3: out = y @ W_proj^T + b_proj  (128 x 1024 x 1024), WMMA tiles.
// ---------------------------------------------------------------------------
__global__ __launch_bounds__(256) void proj_gemm(
    const float* __restrict__ y, const float* __restrict__ W,
    const float* __restrict__ bias, float* __restrict__ out) {
  const int lane = threadIdx.x & 31;
  const int wave = threadIdx.x >> 5;
  const int tile = blockIdx.x * 8 + wave;       // 8 x 64 tiles
  const int rowBase = (tile / 64) * 16;
  const int colBase = (tile % 64) * 16;

  const float* arow = y + (size_t)(rowBase + (lane & 15)) * C_ + (lane >> 4) * 2;
  const float* brow = W + (size_t)(colBase + (lane & 15)) * C_ + (lane >> 4) * 2;

  v8f acc = {};
#pragma unroll 8
  for (int kk = 0; kk < C_; kk += 4) {
    v2f a = *(const v2f*)(arow + kk);
    v2f b = *(const v2f*)(brow + kk);
    acc = wmma_f32_16x16x4(a, b, acc);
  }

  const int n = lane & 15;
  const int mAdd = (lane >> 4) * 8;
#pragma unroll
  for (int i = 0; i < 8; ++i) {
    const int r = rowBase + i + mAdd;
    const int c = colBase + n;
    out[(size_t)r * C_ + c] = acc[i] + bias[c];
  }
}

// ---------------------------------------------------------------------------
extern "C" void kernel_launch(void* const* d_in, const int* in_sizes, int n_in,
                              void* d_out, int out_size, void* d_ws, size_t ws_size,
                              hipStream_t stream) {
  const float* x      = (const float*)d_in[0];
  const float* past_k = (const float*)d_in[1];
  const float* past_v = (const float*)d_in[2];
  const float* W_attn = (const float*)d_in[3];
  const float* b_attn = (const float*)d_in[4];
  const float* W_proj = (const float*)d_in[5];
  const float* b_proj = (const float*)d_in[6];

  float* out   = (float*)d_out;                       // (B,Tn,C) = 131072
  float* out_k = out + (size_t)B_ * Tn * C_;          // (B,H,Tf,HD)
  float* out_v = out_k + (size_t)B_ * H_ * Tf * HD;   // (B,H,Tf,HD)

  float* ws_q = (float*)d_ws;                         // (B,H,Tn,HD) = 131072
  float* ws_y = ws_q + (size_t)B_ * H_ * Tn * HD;     // (B,Tn,C)    = 131072

  // QKV projection: 1536 wave-tiles / 8 waves per block
  qkv_gemm<<<192, 256, 0, stream>>>(x, W_attn, b_attn, ws_q, out_k, out_v);

  // Fused KV-copy + attention: one WG per (b,h), 16 waves, ~148KB dynamic LDS
  const size_t smemBytes =
      (size_t)(8 * Tf + 16 * HD + 16 * 256 + 8) * sizeof(float);
  attn_fused<<<B_ * H_, 512, smemBytes, stream>>>(ws_q, past_k, past_v,
                                                  out_k, out_v, ws_y);

  // Output projection: 512 wave-tiles / 8 waves per block
  proj_gemm<<<64, 256, 0, stream>>>(ws_y, W_proj, b_proj, out);
}